// SGLVRenderer_28372553957950
// MI455X (gfx1250) — compile-verified
//
#include <hip/hip_runtime.h>
#include <math.h>

#define RES_V 16
#define RES_H 32
#define N_SAMP 100
#define GRID_N 256
#define N_CH 11

typedef __attribute__((ext_vector_type(2))) float v2f;
typedef __attribute__((ext_vector_type(8))) float v8f;

__device__ __forceinline__ float rdlane(float x, int lane) {
    return __int_as_float(__builtin_amdgcn_readlane(__float_as_int(x), lane));
}

// Trilinear interp of one channel with zero-outside / clamped-index semantics
// matching the reference's corner() helper. ix/iy/iz are voxel-index coords.
__device__ __forceinline__ float trilerp_ch(const float* __restrict__ volc,
                                            float ix, float iy, float iz) {
    float x0f = floorf(ix), y0f = floorf(iy), z0f = floorf(iz);
    float fx = ix - x0f, fy = iy - y0f, fz = iz - z0f;
    int x0 = (int)x0f, y0 = (int)y0f, z0 = (int)z0f;
    float r = 0.0f;
#pragma unroll
    for (int dz = 0; dz < 2; ++dz) {
#pragma unroll
        for (int dy = 0; dy < 2; ++dy) {
#pragma unroll
            for (int dx = 0; dx < 2; ++dx) {
                int xi = x0 + dx, yi = y0 + dy, zi = z0 + dz;
                bool valid = (xi >= 0) & (xi < GRID_N) & (yi >= 0) & (yi < GRID_N) &
                             (zi >= 0) & (zi < GRID_N);
                int xc = xi < 0 ? 0 : (xi > GRID_N - 1 ? GRID_N - 1 : xi);
                int yc = yi < 0 ? 0 : (yi > GRID_N - 1 ? GRID_N - 1 : yi);
                int zc = zi < 0 ? 0 : (zi > GRID_N - 1 ? GRID_N - 1 : zi);
                long lin = ((long)zc * GRID_N + yc) * GRID_N + xc;
                float w = (dx ? fx : 1.0f - fx) * (dy ? fy : 1.0f - fy) *
                          (dz ? fz : 1.0f - fz);
                r += volc[lin] * (valid ? w : 0.0f);
            }
        }
    }
    return r;
}

// One wave (32 lanes) per direction. Lane L: channel = L&15, sample half = L>>4.
// Weighted channel accumulation runs on the matrix pipe via
// V_WMMA_F32_16X16X4_F32 (A: 16 ch x 4 samples, B: w[k] broadcast over N).
__global__ __launch_bounds__(32)
void sglv_render_kernel(const float* __restrict__ origin,
                        const float* __restrict__ vol,
                        const float* __restrict__ vrange,
                        float* __restrict__ out) {
    const int dirIdx = blockIdx.x;
    const int lane = threadIdx.x;
    const int vI = dirIdx >> 5;
    const int uI = dirIdx & 31;

    const float PI = 3.14159265358979323846f;
    float phi = 2.0f * PI * (float)uI / (float)RES_H;
    float theta = PI * (float)vI / (float)RES_V;
    float st = sinf(theta);
    float dx = st * cosf(phi);
    float dy = cosf(theta);
    float dz = st * sinf(phi);
    float nrm = sqrtf(dx * dx + dy * dy + dz * dz);
    nrm = fmaxf(nrm, 1e-12f);
    dx /= nrm; dy /= nrm; dz /= nrm;

    const float ox = origin[0], oy = origin[1], oz = origin[2];
    const float vr0x = vrange[0], vr0y = vrange[1], vr0z = vrange[2];
    const float vr1x = vrange[3], vr1y = vrange[4], vr1z = vrange[5];

    // Ray-box exactly as reference: t0/t1 = min over axes of positive slab t's.
    const float INF = __int_as_float(0x7f800000);
    float t0 = INF, t1 = INF;
    {
        float d[3]  = {dx, dy, dz};
        float o[3]  = {ox, oy, oz};
        float b0[3] = {vr0x, vr0y, vr0z};
        float b1[3] = {vr1x, vr1y, vr1z};
#pragma unroll
        for (int a = 0; a < 3; ++a) {
            float n0 = b0[a] - o[a], n1 = b1[a] - o[a];
            float tmn, tmx;
            if (d[a] == 0.0f) {
                tmn = n0 > 0.0f ? INF : -INF;
                tmx = n1 > 0.0f ? INF : -INF;
            } else {
                tmn = n0 / d[a];
                tmx = n1 / d[a];
            }
            t0 = fminf(t0, tmn > 0.0f ? tmn : INF);
            t1 = fminf(t1, tmx > 0.0f ? tmx : INF);
        }
    }
    const float t_start = 0.0f;
    const float t_end = fminf(t0, t1);

    // point -> voxel index coordinate scale:
    // ix = ((p - vr0)/(vr1-vr0)*2 - 1 + 1)*0.5*(N-1) = (p - vr0)*(N-1)/(vr1-vr0)
    const float scx = (float)(GRID_N - 1) / (vr1x - vr0x);
    const float scy = (float)(GRID_N - 1) / (vr1y - vr0y);
    const float scz = (float)(GRID_N - 1) / (vr1z - vr0z);

    const int c = lane & 15;      // channel (rows of A), padded 11 -> 16
    const int khalf = lane >> 4;  // 0: K={0,1}, 1: K={2,3}
    const float* volc = vol + (long)(c < N_CH ? c : 0) *
                                  ((long)GRID_N * GRID_N * GRID_N);

    v8f acc = {0.f, 0.f, 0.f, 0.f, 0.f, 0.f, 0.f, 0.f};
    float T = 1.0f;  // running transmittance (wave-uniform)
    const float dt = (t_end - t_start) * (1.0f / (float)(N_SAMP - 1));

    for (int base = 0; base < N_SAMP; base += 4) {
        const int sA = base + (khalf ? 2 : 0);
        const int sB = sA + 1;
        const float tA = t_start + (float)sA * dt;
        const float tB = t_start + (float)sB * dt;

        float aA = 0.0f, aB = 0.0f;
        if (c < N_CH) {
            float px = ox + tA * dx, py = oy + tA * dy, pz = oz + tA * dz;
            aA = trilerp_ch(volc, (px - vr0x) * scx, (py - vr0y) * scy,
                            (pz - vr0z) * scz);
            px = ox + tB * dx; py = oy + tB * dy; pz = oz + tB * dz;
            aB = trilerp_ch(volc, (px - vr0x) * scx, (py - vr0y) * scy,
                            (pz - vr0z) * scz);
        }

        // alpha = channel 3: samples 0,1 live in lane 3; samples 2,3 in lane 19
        const float al0 = rdlane(aA, 3);
        const float al1 = rdlane(aB, 3);
        const float al2 = rdlane(aA, 19);
        const float al3 = rdlane(aB, 19);

        // inclusive cumprod transmittance, weights = alpha * T (wave-uniform)
        T *= (1.0f - al0 + 1e-10f); const float w0 = al0 * T;
        T *= (1.0f - al1 + 1e-10f); const float w1 = al1 * T;
        T *= (1.0f - al2 + 1e-10f); const float w2 = al2 * T;
        T *= (1.0f - al3 + 1e-10f); const float w3 = al3 * T;

        // A (16x4 f32): vgpr0 = K0|K2, vgpr1 = K1|K3 by lane half (ISA 7.12.2)
        v2f A; A.x = aA; A.y = aB;
        // B (4x16 f32): lanes 0-15 hold K=0,1; lanes 16-31 hold K=2,3;
        // each row constant across N -> all D columns equal the channel sums.
        v2f B; B.x = khalf ? w2 : w0; B.y = khalf ? w3 : w1;

        acc = __builtin_amdgcn_wmma_f32_16x16x4_f32(
            false, A, false, B, (short)0, acc, false, false);
    }

    // Extract channel sums from D (vgpr r: M=r @ lanes0-15, M=r+8 @ lanes16-31)
    float accv[11];
#pragma unroll
    for (int i = 0; i < 8; ++i) accv[i] = rdlane(acc[i], 0);
    accv[8]  = rdlane(acc[0], 16);
    accv[9]  = rdlane(acc[1], 16);
    accv[10] = rdlane(acc[2], 16);

    if (lane == 0) {
        const float sdot = dx * accv[8] + dy * accv[9] + dz * accv[10];
        const float e = expf(accv[7] * (sdot - 1.0f));
        out[0 * (RES_V * RES_H) + dirIdx] = accv[0] + accv[4] * e;
        out[1 * (RES_V * RES_H) + dirIdx] = accv[1] + accv[5] * e;
        out[2 * (RES_V * RES_H) + dirIdx] = accv[2] + accv[6] * e;
    }
}

extern "C" void kernel_launch(void* const* d_in, const int* in_sizes, int n_in,
                              void* d_out, int out_size, void* d_ws, size_t ws_size,
                              hipStream_t stream) {
    (void)in_sizes; (void)n_in; (void)out_size; (void)d_ws; (void)ws_size;
    const float* origin = (const float*)d_in[0];
    const float* SGLV   = (const float*)d_in[1];
    const float* vr     = (const float*)d_in[2];
    float* out = (float*)d_out;

    dim3 grid(RES_V * RES_H);  // one wave32 per direction
    dim3 block(32);
    sglv_render_kernel<<<grid, block, 0, stream>>>(origin, SGLV, vr, out);
}